// CrossModalFFTAttn_22393959481696
// MI455X (gfx1250) — compile-verified
//
#include <hip/hip_runtime.h>
#include <hip/hip_bf16.h>

typedef __attribute__((ext_vector_type(2))) float v2f;
typedef __attribute__((ext_vector_type(8))) float v8f;

#define C_IN   128
#define C2     256
#define C4     512
#define H_DIM  192
#define W_DIM  192
#define HW     (H_DIM * W_DIM)      // 36864
#define BATCH  8
#define BHW    (BATCH * HW)         // 294912

// fused tile geometry
#define TW      16
#define TH      8
#define HALO_W  18
#define HALO_H  10
#define NH      (HALO_W * HALO_H)   // 180 real halo cols
#define NT      192                 // padded to 12 N-tiles of 16
#define CP      130                 // padded channel stride (bank-conflict-free, 8B aligned pairs)

// ---------------------------------------------------------------------------
// K1: per-pixel LayerNorm stats over C channels (mean, rstd). One thread/pixel.
// ---------------------------------------------------------------------------
__global__ __launch_bounds__(256) void stats_kernel(const float* __restrict__ src, int C,
                                                    float* __restrict__ mu, float* __restrict__ rs)
{
    int p = blockIdx.x * 256 + threadIdx.x;   // 0..BHW-1
    int b = p / HW, hw = p % HW;
    const float* base = src + (size_t)b * C * HW + hw;
    float s = 0.f, s2 = 0.f;
    for (int c = 0; c < C; ++c) {
        float v = base[(size_t)c * HW];
        s += v; s2 += v * v;
    }
    float m   = s / (float)C;
    float var = s2 / (float)C - m * m;
    mu[p] = m;
    rs[p] = rsqrtf(var + 1e-5f);
}

// ---------------------------------------------------------------------------
// K2: fold LN gamma/beta into conv1x1 weights. One thread per output row.
// ---------------------------------------------------------------------------
__global__ __launch_bounds__(256) void fold_kernel(const float* __restrict__ w, const float* __restrict__ b,
                                                   const float* __restrict__ gam, const float* __restrict__ bet,
                                                   float* __restrict__ wf, float* __restrict__ bf,
                                                   float* __restrict__ rsum, int M)
{
    int m = blockIdx.x * 256 + threadIdx.x;
    if (m >= M) return;
    float bias = b[m], rsm = 0.f;
    for (int c = 0; c < C_IN; ++c) {
        float wv  = w[m * C_IN + c];
        bias     += wv * bet[c];
        float wfv = wv * gam[c];
        wf[m * C_IN + c] = wfv;
        rsm += wfv;
    }
    bf[m] = bias;
    rsum[m] = rsm;
}

// ---------------------------------------------------------------------------
// K3: fused LN + conv1x1 (WMMA f32 16x16x4) + depthwise 3x3 (zero pad).
// Wave w owns M-row block mi = w>>1 and the 6 N-tiles ni = (w&1)*6 + j:
// one A fragment feeds 6 concurrent WMMA accumulators per k-step.
// ---------------------------------------------------------------------------
__global__ __launch_bounds__(256) void fused_ln_conv1x1_dw(
    const float* __restrict__ Xg,
    const float* __restrict__ Wf, const float* __restrict__ Bf, const float* __restrict__ Rsum,
    const float* __restrict__ DW, const float* __restrict__ DB,
    const float* __restrict__ Mu, const float* __restrict__ Rs,
    float* __restrict__ Out, int M)
{
    extern __shared__ float lds[];
    float* Xs  = lds;                 // NT*CP
    float* smu = Xs + NT * CP;        // NT
    float* srs = smu + NT;            // NT
    float* Ck  = srs + NT;            // 64*NT

    const int x0 = blockIdx.x * TW, y0 = blockIdx.y * TH, b = blockIdx.z;
    const int tid = threadIdx.x;

    // halo stats
    for (int t = tid; t < NT; t += 256) {
        int hx = x0 - 1 + (t % HALO_W);
        int hy = y0 - 1 + (t / HALO_W);
        bool in = (t < NH) && hx >= 0 && hx < W_DIM && hy >= 0 && hy < H_DIM;
        int g = in ? (hy * W_DIM + hx) : 0;
        smu[t] = in ? Mu[b * HW + g] : 0.f;
        srs[t] = in ? Rs[b * HW + g] : 0.f;
    }
    // halo input, col-major [col][c]
    for (int idx = tid; idx < NT * C_IN; idx += 256) {
        int t = idx % NT, c = idx / NT;
        int hx = x0 - 1 + (t % HALO_W);
        int hy = y0 - 1 + (t / HALO_W);
        bool in = (t < NH) && hx >= 0 && hx < W_DIM && hy >= 0 && hy < H_DIM;
        Xs[t * CP + c] = in ? Xg[((size_t)b * C_IN + c) * HW + hy * W_DIM + hx] : 0.f;
    }
    __syncthreads();

    const int wave = tid >> 5, lane = tid & 31;
    const int lr = lane & 15, hi = lane >> 4;     // half-wave: K pairs (0,1) | (2,3)
    const int mi = wave >> 1;                     // M-tile, constant per wave
    const int nbase = (wave & 1) * 6;             // first of 6 N-tiles
    const int nchunks = M / 64;

    for (int ch = 0; ch < nchunks; ++ch) {
        const int mbase = ch * 64;
        const int m0 = mbase + mi * 16;
        const float* Arow = Wf + (size_t)(m0 + lr) * C_IN;

        // per-chunk epilogue constants (8 consecutive floats each)
        float rsv[8], bfv[8];
        #pragma unroll
        for (int i = 0; i < 8; ++i) {
            rsv[i] = Rsum[m0 + hi * 8 + i];
            bfv[i] = Bf[m0 + hi * 8 + i];
        }

        v8f acc[6];
        #pragma unroll
        for (int j = 0; j < 6; ++j) acc[j] = (v8f){};

        // software-pipelined GEMM: 1 A-frag + 6 B-frags feed 6 WMMAs per step
        v2f a_c, b_c[6];
        {
            a_c.x = Arow[hi * 2]; a_c.y = Arow[hi * 2 + 1];
            #pragma unroll
            for (int j = 0; j < 6; ++j) {
                const float* Bcol = Xs + ((nbase + j) * 16 + lr) * CP;
                b_c[j].x = Bcol[hi * 2]; b_c[j].y = Bcol[hi * 2 + 1];
            }
        }
        #pragma unroll
        for (int k = 0; k < C_IN; k += 4) {
            v2f a_n = a_c, b_n[6];
            #pragma unroll
            for (int j = 0; j < 6; ++j) b_n[j] = b_c[j];
            if (k + 4 < C_IN) {
                int ka = k + 4 + hi * 2;
                a_n.x = Arow[ka]; a_n.y = Arow[ka + 1];
                #pragma unroll
                for (int j = 0; j < 6; ++j) {
                    const float* Bcol = Xs + ((nbase + j) * 16 + lr) * CP;
                    b_n[j].x = Bcol[ka]; b_n[j].y = Bcol[ka + 1];
                }
            }
            #pragma unroll
            for (int j = 0; j < 6; ++j)
                acc[j] = __builtin_amdgcn_wmma_f32_16x16x4_f32(
                            false, a_c, false, b_c[j], (short)0, acc[j], false, false);
            a_c = a_n;
            #pragma unroll
            for (int j = 0; j < 6; ++j) b_c[j] = b_n[j];
        }

        // LN epilogue -> chunk buffer
        #pragma unroll
        for (int j = 0; j < 6; ++j) {
            int n = (nbase + j) * 16 + lr;
            float mu_n = smu[n], rs_n = srs[n];
            #pragma unroll
            for (int i = 0; i < 8; ++i) {
                int ml = mi * 16 + hi * 8 + i;        // row within chunk
                Ck[ml * NT + n] = rs_n * (acc[j][i] - mu_n * rsv[i]) + bfv[i];
            }
        }
        __syncthreads();

        // depthwise 3x3 over the 16x8 output tile for this 64-channel chunk
        for (int idx = tid; idx < 64 * TW * TH; idx += 256) {
            int p = idx % (TW * TH), cch = idx / (TW * TH);
            int px = p % TW, py = p / TW;
            int gx = x0 + px, gy = y0 + py;
            int m = mbase + cch;
            const float* dwp = DW + m * 9;
            float a = DB[m];
            #pragma unroll
            for (int dy = -1; dy <= 1; ++dy)
                #pragma unroll
                for (int dx = -1; dx <= 1; ++dx) {
                    int ny = gy + dy, nx = gx + dx;
                    float v = (ny >= 0 && ny < H_DIM && nx >= 0 && nx < W_DIM)
                            ? Ck[cch * NT + (py + dy + 1) * HALO_W + (px + dx + 1)] : 0.f;
                    a += dwp[(dy + 1) * 3 + (dx + 1)] * v;
                }
            Out[((size_t)b * M + m) * HW + gy * W_DIM + gx] = a;
        }
        __syncthreads();
    }
}

// ---------------------------------------------------------------------------
// K4: per-8x8-patch circular convolution corr = q (*) k, in place into Q.
// ---------------------------------------------------------------------------
__global__ __launch_bounds__(256) void patch_corr(float* __restrict__ Q, const float* __restrict__ KV)
{
    __shared__ float sq[4][64], sk[4][64];
    int tid = threadIdx.x;
    int pp = tid >> 6, o = tid & 63;
    long pc    = (long)blockIdx.x * 4 + pp;
    int patch  = (int)(pc % 576);
    int chb    = (int)(pc / 576);
    int chn    = chb % C2;
    int b      = chb / C2;
    int px = (patch % 24) * 8, py = (patch / 24) * 8;
    size_t qoff = ((size_t)(b * C2 + chn)) * HW + py * W_DIM + px;
    size_t koff = ((size_t)(b * C4 + chn)) * HW + py * W_DIM + px;  // k = first 256 ch of kv
    int r = o >> 3, cc = o & 7;
    sq[pp][o] = Q[qoff + r * W_DIM + cc];
    sk[pp][o] = KV[koff + r * W_DIM + cc];
    __syncthreads();
    int om = o >> 3, on = o & 7;
    float acc = 0.f;
    #pragma unroll
    for (int s = 0; s < 64; ++s) {
        int sm = s >> 3, sn = s & 7;
        acc += sq[pp][s] * sk[pp][((om - sm) & 7) * 8 + ((on - sn) & 7)];
    }
    Q[qoff + om * W_DIM + on] = acc;
}

// ---------------------------------------------------------------------------
// K5: final projection: out = proj_w @ (v .* corr_ln) + proj_b  (WMMA f32).
// Wave w owns M-tile mi = w; 4 concurrent N-tile accumulators share each A-frag.
// ---------------------------------------------------------------------------
#define KP 258
__global__ __launch_bounds__(256) void final_proj(
    const float* __restrict__ Corr, const float* __restrict__ KV,
    const float* __restrict__ MuC, const float* __restrict__ RsC,
    const float* __restrict__ LW, const float* __restrict__ LB,
    const float* __restrict__ PW, const float* __restrict__ PB,
    float* __restrict__ Out)
{
    extern __shared__ float lds[];
    float* Bt = lds;                       // [64][KP]
    const int b = blockIdx.y;
    const int n0g = blockIdx.x * 64;
    const int tid = threadIdx.x;

    {   // build Bt = v .* corr_ln, column-major [col][c]
        int colt = tid & 63, c0 = tid >> 6;
        int n = n0g + colt;
        float mu_n = MuC[b * HW + n], rs_n = RsC[b * HW + n];
        for (int it = 0; it < 64; ++it) {
            int c = c0 + it * 4;
            float corr = Corr[((size_t)(b * C2 + c)) * HW + n];
            float vv   = KV[((size_t)(b * C4 + C2 + c)) * HW + n];
            Bt[colt * KP + c] = vv * ((corr - mu_n) * rs_n * LW[c] + LB[c]);
        }
    }
    __syncthreads();

    const int wave = tid >> 5, lane = tid & 31;
    const int lr = lane & 15, hi = lane >> 4;
    const int m0 = wave * 16;                          // M-tile per wave
    const float* Arow = PW + (size_t)(m0 + lr) * C2;

    v8f acc[4];
    #pragma unroll
    for (int j = 0; j < 4; ++j) acc[j] = (v8f){};

    v2f a_c, b_c[4];
    {
        a_c.x = Arow[hi * 2]; a_c.y = Arow[hi * 2 + 1];
        #pragma unroll
        for (int j = 0; j < 4; ++j) {
            const float* Bcol = Bt + (j * 16 + lr) * KP;
            b_c[j].x = Bcol[hi * 2]; b_c[j].y = Bcol[hi * 2 + 1];
        }
    }
    #pragma unroll
    for (int k = 0; k < C2; k += 4) {
        v2f a_n = a_c, b_n[4];
        #pragma unroll
        for (int j = 0; j < 4; ++j) b_n[j] = b_c[j];
        if (k + 4 < C2) {
            int ka = k + 4 + hi * 2;
            a_n.x = Arow[ka]; a_n.y = Arow[ka + 1];
            #pragma unroll
            for (int j = 0; j < 4; ++j) {
                const float* Bcol = Bt + (j * 16 + lr) * KP;
                b_n[j].x = Bcol[ka]; b_n[j].y = Bcol[ka + 1];
            }
        }
        #pragma unroll
        for (int j = 0; j < 4; ++j)
            acc[j] = __builtin_amdgcn_wmma_f32_16x16x4_f32(
                        false, a_c, false, b_c[j], (short)0, acc[j], false, false);
        a_c = a_n;
        #pragma unroll
        for (int j = 0; j < 4; ++j) b_c[j] = b_n[j];
    }

    float pbv[8];
    #pragma unroll
    for (int i = 0; i < 8; ++i) pbv[i] = PB[m0 + hi * 8 + i];
    #pragma unroll
    for (int j = 0; j < 4; ++j) {
        int n = n0g + j * 16 + lr;
        #pragma unroll
        for (int i = 0; i < 8; ++i) {
            int m = m0 + i + hi * 8;
            Out[((size_t)b * C_IN + m) * HW + n] = acc[j][i] + pbv[i];
        }
    }
}

// ---------------------------------------------------------------------------
extern "C" void kernel_launch(void* const* d_in, const int* in_sizes, int n_in,
                              void* d_out, int out_size, void* d_ws, size_t ws_size,
                              hipStream_t stream)
{
    const float* x        = (const float*)d_in[0];
    const float* evt      = (const float*)d_in[1];
    const float* ln_img_w = (const float*)d_in[2];
    const float* ln_img_b = (const float*)d_in[3];
    const float* ln_evt_w = (const float*)d_in[4];
    const float* ln_evt_b = (const float*)d_in[5];
    const float* q_w      = (const float*)d_in[6];
    const float* q_b      = (const float*)d_in[7];
    const float* q_dw_w   = (const float*)d_in[8];
    const float* q_dw_b   = (const float*)d_in[9];
    const float* kv_w     = (const float*)d_in[10];
    const float* kv_b     = (const float*)d_in[11];
    const float* kv_dw_w  = (const float*)d_in[12];
    const float* kv_dw_b  = (const float*)d_in[13];
    const float* ln_c_w   = (const float*)d_in[14];
    const float* ln_c_b   = (const float*)d_in[15];
    const float* proj_w   = (const float*)d_in[16];
    const float* proj_b   = (const float*)d_in[17];
    float* out = (float*)d_out;

    // workspace carve-up (floats)
    float* ws = (float*)d_ws;
    size_t off = 0;
    float* qbuf  = ws + off; off += (size_t)BATCH * C2 * HW;   // q, later corr (in place)
    float* kvbuf = ws + off; off += (size_t)BATCH * C4 * HW;   // k (0..255) | v (256..511)
    float* mu_x  = ws + off; off += BHW;
    float* rs_x  = ws + off; off += BHW;
    float* mu_e  = ws + off; off += BHW;
    float* rs_e  = ws + off; off += BHW;
    float* mu_c  = ws + off; off += BHW;
    float* rs_c  = ws + off; off += BHW;
    float* qwf   = ws + off; off += (size_t)C2 * C_IN;
    float* qbf   = ws + off; off += C2;
    float* qrs   = ws + off; off += C2;
    float* kvwf  = ws + off; off += (size_t)C4 * C_IN;
    float* kvbf  = ws + off; off += C4;
    float* kvrs  = ws + off; off += C4;

    const size_t fused_lds = (size_t)(NT * CP + 2 * NT + 64 * NT) * sizeof(float); // ~147 KB
    const size_t proj_lds  = (size_t)(64 * KP) * sizeof(float);                    // ~66 KB

    // 1) LN stats for x and evt
    stats_kernel<<<BHW / 256, 256, 0, stream>>>(x,   C_IN, mu_x, rs_x);
    stats_kernel<<<BHW / 256, 256, 0, stream>>>(evt, C_IN, mu_e, rs_e);

    // 2) fold LN gamma/beta into conv weights
    fold_kernel<<<1, 256, 0, stream>>>(q_w,  q_b,  ln_img_w, ln_img_b, qwf,  qbf,  qrs,  C2);
    fold_kernel<<<2, 256, 0, stream>>>(kv_w, kv_b, ln_evt_w, ln_evt_b, kvwf, kvbf, kvrs, C4);

    // 3) fused LN + conv1x1 + dwconv3x3 for q and kv
    dim3 fgrid(W_DIM / TW, H_DIM / TH, BATCH);
    fused_ln_conv1x1_dw<<<fgrid, 256, fused_lds, stream>>>(
        x,   qwf,  qbf,  qrs,  q_dw_w,  q_dw_b,  mu_x, rs_x, qbuf,  C2);
    fused_ln_conv1x1_dw<<<fgrid, 256, fused_lds, stream>>>(
        evt, kvwf, kvbf, kvrs, kv_dw_w, kv_dw_b, mu_e, rs_e, kvbuf, C4);

    // 4) per-patch circular convolution (FFT product), in place into qbuf
    patch_corr<<<(BATCH * C2 * 576) / 4, 256, 0, stream>>>(qbuf, kvbuf);

    // 5) corr LN stats over 256 channels
    stats_kernel<<<BHW / 256, 256, 0, stream>>>(qbuf, C2, mu_c, rs_c);

    // 6) final projection GEMM with fused corr-LN and v multiply
    dim3 pgrid(HW / 64, BATCH);
    final_proj<<<pgrid, 256, proj_lds, stream>>>(
        qbuf, kvbuf, mu_c, rs_c, ln_c_w, ln_c_b, proj_w, proj_b, out);
    (void)in_sizes; (void)n_in; (void)out_size; (void)ws_size;
}